// KNearestNeigbors_58617713656403
// MI455X (gfx1250) — compile-verified
//
#include <hip/hip_runtime.h>
#include <hip/hip_bf16.h>
#include <math.h>
#include <stdint.h>

typedef __attribute__((ext_vector_type(2))) float v2f;
typedef __attribute__((ext_vector_type(8))) float v8f;

#define NEG_INF (-3.402823466e+38f)
#define TOPK_CHUNK 2752
#define ROW_STRIDE 132   // 128 + 4 pad floats -> conflict-free ds_load_b64 A reads

// ---------------------------------------------------------------------------
// Kernel 1: normalize the query vector (D=128). qn = q / sqrt(sum(q^2)+1e-12)
// ---------------------------------------------------------------------------
__global__ __launch_bounds__(128) void qnorm_kernel(const float* __restrict__ emb,
                                                    float* __restrict__ qn) {
  __shared__ float s[128];
  const int t = threadIdx.x;
  float v = emb[t];
  s[t] = v * v;
  __syncthreads();
  for (int st = 64; st > 0; st >>= 1) {
    if (t < st) s[t] += s[t + st];
    __syncthreads();
  }
  float inv = 1.0f / sqrtf(s[0] + 1e-12f);
  qn[t] = v * inv;
}

// ---------------------------------------------------------------------------
// Kernel 2: fused cosine-sim (V_WMMA_F32_16X16X4_F32) + per-block top-11.
// 128 threads = 4 waves; each wave owns 16 rows staged in LDS via
// GLOBAL_LOAD_ASYNC_TO_LDS_B128 (ASYNCcnt), avoiding the VGPR round-trip.
// ---------------------------------------------------------------------------
__global__ __launch_bounds__(128) void knn_score_topk_kernel(
    const float* __restrict__ coll, const float* __restrict__ qn, int nrows,
    float* __restrict__ cand_val, int* __restrict__ cand_idx) {
  __shared__ __align__(16) float tiles[4 * 16 * ROW_STRIDE];  // 33,792 B
  __shared__ __align__(16) float qs[128];
  __shared__ float blkscores[64];
  __shared__ float redv[4];
  __shared__ int redi[4];
  __shared__ int widx_s;

  const int t = threadIdx.x;
  const int warp = t >> 5;
  const int lane = t & 31;
  const int half = lane >> 4;   // 0: lanes 0-15, 1: lanes 16-31
  const int mrow = lane & 15;

  const int rowbase = blockIdx.x * 64 + warp * 16;
  float* tile = &tiles[warp * (16 * ROW_STRIDE)];

  // Async-stage 16 rows directly into LDS: one 512B row per iteration
  // (each lane copies 16B). LDS offset = low 32 bits of the shared address.
  const uint32_t lane_lds =
      (uint32_t)(uintptr_t)(&tile[0]) + (uint32_t)lane * 16u;
  #pragma unroll
  for (int r = 0; r < 16; ++r) {
    const int row = rowbase + r;              // wave-uniform condition
    if (row < nrows) {
      const uint32_t ldsa = lane_lds + (uint32_t)(r * ROW_STRIDE * 4);
      const uint32_t goff = (uint32_t)row * 512u + (uint32_t)lane * 16u;
      asm volatile("global_load_async_to_lds_b128 %0, %1, %2"
                   :
                   : "v"(ldsa), "v"(goff), "s"(coll)
                   : "memory");
    } else {
      *reinterpret_cast<float4*>(&tile[r * ROW_STRIDE + lane * 4]) =
          make_float4(0.f, 0.f, 0.f, 0.f);
    }
  }
  qs[t] = qn[t];
  asm volatile("s_wait_asynccnt 0x0" ::: "memory");
  __syncthreads();  // qs visible block-wide; async tile writes complete

  // Row half-norms: 2 lanes per row, 64 floats each, then fold halves
  float sq = 0.f;
  #pragma unroll
  for (int j = 0; j < 16; ++j) {
    const float* p = &tile[mrow * ROW_STRIDE + half * 64 + j * 4];
    sq = fmaf(p[0], p[0], fmaf(p[1], p[1], fmaf(p[2], p[2], fmaf(p[3], p[3], sq))));
  }
  const float nrm = sq + __shfl_xor(sq, 16);  // lane l holds ||row (l&15)||^2

  // Dot products via fp32 WMMA: A = 16x4 row slice, B = 4x16 broadcast q.
  // f32 A layout: lanes 0-15 -> K=k,k+1 ; lanes 16-31 -> K=k+2,k+3 (koff=2).
  v8f c = {0.f, 0.f, 0.f, 0.f, 0.f, 0.f, 0.f, 0.f};
  const int koff = half * 2;
  #pragma unroll
  for (int k = 0; k < 128; k += 4) {
    v2f a = *reinterpret_cast<const v2f*>(&tile[mrow * ROW_STRIDE + k + koff]);
    v2f b = *reinterpret_cast<const v2f*>(&qs[k + koff]);
    c = __builtin_amdgcn_wmma_f32_16x16x4_f32(false, a, false, b, (short)0, c,
                                              false, false);
  }

  // C/D layout: VGPR v -> row v (lanes 0-15) / row v+8 (lanes 16-31); all N equal.
  #pragma unroll
  for (int v = 0; v < 8; ++v) {
    const int rm = v + half * 8;
    const float nr = __shfl(nrm, rm);  // norm of row rm lives on lane rm
    const float sc = c[v] * (1.0f / sqrtf(nr + 1e-12f));
    if (mrow == v) {
      const int grow = rowbase + rm;
      blkscores[warp * 16 + rm] = (grow < nrows) ? sc : NEG_INF;
    }
  }
  __syncthreads();

  // Block-level top-11 over the 64 scores (parallel masked argmax)
  float myv = (t < 64) ? blkscores[t] : NEG_INF;
  for (int r = 0; r < 11; ++r) {
    float bv = myv;
    int bi = t;
    #pragma unroll
    for (int m = 16; m > 0; m >>= 1) {
      float ov = __shfl_xor(bv, m);
      int oi = __shfl_xor(bi, m);
      if (ov > bv || (ov == bv && oi < bi)) { bv = ov; bi = oi; }
    }
    if (lane == 0) { redv[warp] = bv; redi[warp] = bi; }
    __syncthreads();
    if (t == 0) {
      float fbv = redv[0]; int fbi = redi[0];
      for (int w = 1; w < 4; ++w)
        if (redv[w] > fbv || (redv[w] == fbv && redi[w] < fbi)) {
          fbv = redv[w]; fbi = redi[w];
        }
      widx_s = fbi;
      cand_val[blockIdx.x * 11 + r] = fbv;
      cand_idx[blockIdx.x * 11 + r] = blockIdx.x * 64 + fbi;
    }
    __syncthreads();
    if (t == widx_s) myv = NEG_INF;
  }
}

// ---------------------------------------------------------------------------
// Kernel 3/4: generic top-11 reduction. Block b handles [b*chunk, b*chunk+cnt),
// stages into LDS, 11 rounds of masked argmax, outputs sorted-desc 11 pairs.
// ---------------------------------------------------------------------------
__global__ __launch_bounds__(256) void topk11_kernel(
    const float* __restrict__ ivals, const int* __restrict__ iidx, int n,
    int chunk, float* __restrict__ ovals, int* __restrict__ oidx) {
  __shared__ float cv[TOPK_CHUNK];
  __shared__ int ci[TOPK_CHUNK];
  __shared__ float redv[8];
  __shared__ int redi[8];
  __shared__ int redp[8];

  const int t = threadIdx.x;
  const int start = blockIdx.x * chunk;
  int cnt = n - start;
  if (cnt > chunk) cnt = chunk;
  if (cnt < 0) cnt = 0;

  for (int i = t; i < cnt; i += 256) { cv[i] = ivals[start + i]; ci[i] = iidx[start + i]; }
  __syncthreads();

  for (int r = 0; r < 11; ++r) {
    float bv = NEG_INF;
    int bidx = 0x7fffffff;
    int bp = -1;
    for (int i = t; i < cnt; i += 256) {
      float v = cv[i];
      int id = ci[i];
      if (v > bv || (v == bv && id < bidx)) { bv = v; bidx = id; bp = i; }
    }
    #pragma unroll
    for (int m = 16; m > 0; m >>= 1) {
      float ov = __shfl_xor(bv, m);
      int oi = __shfl_xor(bidx, m);
      int op = __shfl_xor(bp, m);
      if (ov > bv || (ov == bv && oi < bidx)) { bv = ov; bidx = oi; bp = op; }
    }
    const int w = t >> 5;
    if ((t & 31) == 0) { redv[w] = bv; redi[w] = bidx; redp[w] = bp; }
    __syncthreads();
    if (t == 0) {
      float fbv = redv[0]; int fbi = redi[0]; int fbp = redp[0];
      for (int k = 1; k < 8; ++k)
        if (redv[k] > fbv || (redv[k] == fbv && redi[k] < fbi)) {
          fbv = redv[k]; fbi = redi[k]; fbp = redp[k];
        }
      ovals[blockIdx.x * 11 + r] = fbv;
      oidx[blockIdx.x * 11 + r] = fbi;
      if (fbp >= 0) cv[fbp] = NEG_INF;  // mask winner
    }
    __syncthreads();
  }
}

// ---------------------------------------------------------------------------
// Kernel 5: classification from global top-11 (ranks 1..9 per reference quirk)
// ---------------------------------------------------------------------------
__global__ void classify_kernel(const float* __restrict__ fv,
                                const int* __restrict__ fi,
                                const int* __restrict__ labels, int nrows,
                                float* __restrict__ out) {
  if (threadIdx.x != 0 || blockIdx.x != 0) return;
  int labs[9];
  float probs[9];
  for (int j = 0; j < 9; ++j) {
    int idx = fi[1 + j];
    if (idx < 0) idx = 0;
    if (idx >= nrows) idx = nrows - 1;
    labs[j] = labels[idx];
    probs[j] = fv[1 + j];
  }
  int best_count = -1, best_class = 0x7fffffff;
  for (int j = 0; j < 9; ++j) {
    int cjs = 0;
    for (int k = 0; k < 9; ++k) cjs += (labs[k] == labs[j]);
    if (cjs > best_count || (cjs == best_count && labs[j] < best_class)) {
      best_count = cjs;
      best_class = labs[j];
    }
  }
  int first = 0;
  for (int j = 8; j >= 0; --j)
    if (labs[j] == best_class) first = j;
  out[0] = (float)best_class;
  out[1] = probs[first];
  out[2] = (float)best_count / 9.0f;
}

// ---------------------------------------------------------------------------
extern "C" void kernel_launch(void* const* d_in, const int* in_sizes, int n_in,
                              void* d_out, int out_size, void* d_ws,
                              size_t ws_size, hipStream_t stream) {
  const float* emb = (const float*)d_in[0];     // (1,128) f32
  const float* coll = (const float*)d_in[1];    // (N,128) f32
  const int* labels = (const int*)d_in[2];      // (N,) int
  const int nrows = in_sizes[1] / 128;

  // workspace carve (all 4-byte aligned)
  float* qn = (float*)d_ws;                     // 128
  const int blocksB = (nrows + 63) / 64;
  const int candN = blocksB * 11;
  float* cand_val = qn + 128;
  int* cand_idx = (int*)(cand_val + candN);
  const int chunk = 2688;
  const int B1 = (candN + chunk - 1) / chunk;   // 64 for N=1e6
  const int n2 = B1 * 11;
  float* c2v = (float*)(cand_idx + candN);
  int* c2i = (int*)(c2v + n2);
  float* fv = (float*)(c2i + n2);
  int* fi = (int*)(fv + 11);

  qnorm_kernel<<<1, 128, 0, stream>>>(emb, qn);
  knn_score_topk_kernel<<<blocksB, 128, 0, stream>>>(coll, qn, nrows, cand_val,
                                                     cand_idx);
  topk11_kernel<<<B1, 256, 0, stream>>>(cand_val, cand_idx, candN, chunk, c2v,
                                        c2i);
  topk11_kernel<<<1, 256, 0, stream>>>(c2v, c2i, n2, TOPK_CHUNK, fv, fi);
  classify_kernel<<<1, 32, 0, stream>>>(fv, fi, labels, nrows, (float*)d_out);
}